// GaussianLSTMCore_7868380086565
// MI455X (gfx1250) — compile-verified
//
#include <hip/hip_runtime.h>
#include <math.h>

// ---------------------------------------------------------------------------
// GaussianLSTMCore for MI455X (gfx1250, wave32, WMMA).
//
// Kernel 1: x_proj = x @ W_ih + bias (68.7 GFLOP) via V_WMMA_F32_16X16X4_F32.
// Kernel 2: persistent scan, 32 WGs = 2 row-groups (16 rows, WMMA M=16)
//           x 16 hidden-slice groups (32 hidden units).  Row-groups are fully
//           independent -> per-row-group 16-WG software barriers (2 per step).
//           c state lives in LDS for the whole scan; h_reparam slice lives in
//           LDS transposed (K-major, pad->conflict-free b64 B-operand loads).
// ---------------------------------------------------------------------------

#define B_   32
#define S_   2048
#define I_   256
#define H_   512
#define G4_  2048   // 4*H
#define H2_  1024   // 2*H
#define NWG  32     // persistent workgroups in the scan
#define RGWG 16     // workgroups per row-group barrier

typedef __attribute__((ext_vector_type(2))) float v2f;
typedef __attribute__((ext_vector_type(8))) float v8f;

#define WMMA_F32X4(A, Bv, C) \
  __builtin_amdgcn_wmma_f32_16x16x4_f32(false, (A), false, (Bv), (short)0, (C), false, false)

__device__ __forceinline__ float sigmoidf_(float x) {
  return 1.0f / (1.0f + __expf(-x));
}
__device__ __forceinline__ float softplusf_(float x) {
  return (x > 20.0f) ? x : log1pf(__expf(x));
}

// Sense/epoch barrier across the RGWG workgroups of one row-group.
__device__ __forceinline__ void rg_barrier(unsigned* cnt, unsigned* epoch,
                                           unsigned target) {
  __threadfence();
  __syncthreads();
  if (threadIdx.x == 0) {
    unsigned prev = __hip_atomic_fetch_add(cnt, 1u, __ATOMIC_ACQ_REL,
                                           __HIP_MEMORY_SCOPE_AGENT);
    if (prev == RGWG - 1u) {
      __hip_atomic_store(cnt, 0u, __ATOMIC_RELAXED, __HIP_MEMORY_SCOPE_AGENT);
      __hip_atomic_fetch_add(epoch, 1u, __ATOMIC_ACQ_REL,
                             __HIP_MEMORY_SCOPE_AGENT);
    } else {
      while (__hip_atomic_load(epoch, __ATOMIC_ACQUIRE,
                               __HIP_MEMORY_SCOPE_AGENT) < target) {
        __builtin_amdgcn_s_sleep(1);
      }
    }
  }
  __syncthreads();
}

// ---------------------------------------------------------------------------
// Kernel 0: zero h0 and the barrier words (ws is poisoned by the harness).
// ---------------------------------------------------------------------------
__global__ void init_state(float* hbuf, unsigned* sync) {
  int i = blockIdx.x * blockDim.x + threadIdx.x;
  if (i < B_ * H_) hbuf[i] = 0.0f;
  if (i < 256) sync[i] = 0u;
}

// ---------------------------------------------------------------------------
// Kernel 1: x_proj[m, n] = sum_k x[m,k] * Wih[k,n] + bias[n],  m = b*S + s.
// ---------------------------------------------------------------------------
__global__ __launch_bounds__(128, 1)
void xproj_gemm(const float* __restrict__ X, const float* __restrict__ Wih,
                const float* __restrict__ bias, float* __restrict__ xproj) {
  const int tid  = threadIdx.x;
  const int lane = tid & 31;
  const int wave = tid >> 5;
  const int lrow = lane & 15;
  const int lhi  = lane >> 4;
  const int m0   = blockIdx.x * 16;
  const int col  = blockIdx.y * 64 + wave * 16 + lrow;

  const float* xrow = &X[(size_t)(m0 + lrow) * I_];
  v8f acc = {};
  for (int k0 = 0; k0 < I_; k0 += 4) {
    const int ka = k0 + 2 * lhi;
    v2f a = *(const v2f*)&xrow[ka];
    v2f b = { Wih[(size_t)ka * G4_ + col],
              Wih[(size_t)(ka + 1) * G4_ + col] };
    acc = WMMA_F32X4(a, b, acc);
  }
  const float bv = bias[col];
  for (int v = 0; v < 8; ++v) {
    const int m = m0 + v + 8 * lhi;
    xproj[(size_t)m * G4_ + col] = acc[v] + bv;
  }
}

// ---------------------------------------------------------------------------
// Kernel 2: persistent scan.  rg = bx>>4 (16 batch rows), cg = bx&15
// (32 hidden units).  128 threads = 4 waves.
// ---------------------------------------------------------------------------
template <bool USE_XPROJ>
__global__ __launch_bounds__(128, 1)
void lstm_scan(const float* __restrict__ X,    const float* __restrict__ Wih,
               const float* __restrict__ Whh,  const float* __restrict__ bias,
               const float* __restrict__ Hrep, const float* __restrict__ eps,
               const float* __restrict__ xproj,
               float* __restrict__ hbuf, float* __restrict__ prehbuf,
               unsigned* syncw,
               float* __restrict__ hseq, float* __restrict__ cseq,
               float* __restrict__ mus,  float* __restrict__ stds,
               float* __restrict__ hT,   float* __restrict__ cT) {
  __shared__ float sh[16 * 516];                  // h / pre_h row block (padded)
  __shared__ float sc[16 * 32];                   // persistent c state
  __shared__ float sg[4 * 16 * 32];               // gate / param staging
  __shared__ float shr[64 * 516];                 // h_reparam slice, K-major (transposed)
  __shared__ float sx[USE_XPROJ ? 1 : 16 * 260];  // x_t block (fused mode only)

  const int tid  = threadIdx.x;
  const int lane = tid & 31;
  const int wave = tid >> 5;
  const int lrow = lane & 15;
  const int lhi  = lane >> 4;
  const int rg   = blockIdx.x >> 4;     // 0..1
  const int cg   = blockIdx.x & 15;     // 0..15
  const int j0   = cg * 32;             // first hidden unit of our slice
  const int r0   = rg * 16;             // first batch row

  unsigned* cnt   = syncw + rg * 64;    // per-row-group barrier words
  unsigned* epoch = syncw + rg * 64 + 32;

  // Gate-tile assignment: tile tt in 0..7 -> gate g = tt>>1, sub-cols (tt&1)*16.
  const int ttA = wave,     gA = ttA >> 1, sA = ttA & 1;
  const int ttB = wave + 4, gB = ttB >> 1, sB = ttB & 1;
  const int colA = gA * H_ + j0 + sA * 16 + lrow;
  const int colB = gB * H_ + j0 + sB * 16 + lrow;

  for (int i = tid; i < 16 * 32; i += 128) sc[i] = 0.0f;   // c0 = 0

  // Stage h_reparam slice into LDS once, transposed to K-major:
  // local col lc 0..31 -> mu col j0+lc ; lc 32..63 -> std col H_+j0+(lc-32).
  for (int i = tid; i < 512 * 64; i += 128) {
    const int k = i >> 6, lc = i & 63;
    const int gcol = (lc < 32) ? (j0 + lc) : (H_ + j0 + (lc - 32));
    shr[lc * 516 + k] = Hrep[(size_t)k * H2_ + gcol];
  }

  unsigned target = 0;
  for (int t = 0; t < S_; ++t) {
    const bool last = (t == S_ - 1);

    // -------- stage h_{t-1} block [16 x 512] into LDS --------
    for (int i = tid * 4; i < 16 * 512; i += 128 * 4) {
      const int r = i >> 9, c = i & 511;
      const float4 v = *(const float4*)&hbuf[(r0 + r) * H_ + c];
      float* d = &sh[r * 516 + c];
      d[0] = v.x; d[1] = v.y; d[2] = v.z; d[3] = v.w;
    }
    if constexpr (!USE_XPROJ) {
      for (int i = tid * 4; i < 16 * 256; i += 128 * 4) {
        const int r = i >> 8, c = i & 255;
        const float4 v = *(const float4*)&X[((size_t)(r0 + r) * S_ + t) * I_ + c];
        float* d = &sx[r * 260 + c];
        d[0] = v.x; d[1] = v.y; d[2] = v.z; d[3] = v.w;
      }
    }
    // Prefetch next step's streamed tiles (global_prefetch_b8) while we compute.
    if (t + 1 < S_) {
      if constexpr (USE_XPROJ) {
        const int r = tid >> 3, sub = tid & 7;            // 16 rows x 4 gates x 2 halves
        const int g = sub >> 1, half = (sub & 1) * 16;
        __builtin_prefetch(
            &xproj[((size_t)(r0 + r) * S_ + (t + 1)) * G4_ + g * H_ + j0 + half], 0, 1);
      }
      if (tid < 32) {                                      // eps: 16 rows x 2 lines
        const int r = tid >> 1, half = (tid & 1) * 16;
        __builtin_prefetch(&eps[((size_t)(r0 + r) * S_ + (t + 1)) * H_ + j0 + half], 0, 1);
      }
    }
    __syncthreads();

    // -------- phase 1: gates = xp_t + h @ W_hh --------
    v8f acc0 = {}, acc1 = {};
    if constexpr (!USE_XPROJ) {
      for (int k0 = 0; k0 < I_; k0 += 4) {
        const int ka = k0 + 2 * lhi;
        v2f a  = *(const v2f*)&sx[lrow * 260 + ka];
        v2f b0 = { Wih[(size_t)ka * G4_ + colA], Wih[(size_t)(ka + 1) * G4_ + colA] };
        acc0 = WMMA_F32X4(a, b0, acc0);
        v2f b1 = { Wih[(size_t)ka * G4_ + colB], Wih[(size_t)(ka + 1) * G4_ + colB] };
        acc1 = WMMA_F32X4(a, b1, acc1);
      }
    }
    for (int k0 = 0; k0 < H_; k0 += 4) {
      const int ka = k0 + 2 * lhi;
      v2f a  = *(const v2f*)&sh[lrow * 516 + ka];
      v2f b0 = { Whh[(size_t)ka * G4_ + colA], Whh[(size_t)(ka + 1) * G4_ + colA] };
      acc0 = WMMA_F32X4(a, b0, acc0);
      v2f b1 = { Whh[(size_t)ka * G4_ + colB], Whh[(size_t)(ka + 1) * G4_ + colB] };
      acc1 = WMMA_F32X4(a, b1, acc1);
    }
    for (int v = 0; v < 8; ++v) {
      const int m = v + 8 * lhi;
      float g0, g1;
      if constexpr (USE_XPROJ) {
        g0 = acc0[v] + xproj[((size_t)(r0 + m) * S_ + t) * G4_ + colA];
        g1 = acc1[v] + xproj[((size_t)(r0 + m) * S_ + t) * G4_ + colB];
      } else {
        g0 = acc0[v] + bias[colA];
        g1 = acc1[v] + bias[colB];
      }
      sg[(gA * 16 + m) * 32 + sA * 16 + lrow] = g0;
      sg[(gB * 16 + m) * 32 + sB * 16 + lrow] = g1;
    }
    __syncthreads();

    // -------- gate nonlinearities, c update, pre_h --------
    for (int u = tid; u < 512; u += 128) {
      const int m = u >> 5, c = u & 31;
      const float it = sigmoidf_(sg[(0 * 16 + m) * 32 + c]);
      const float ft = sigmoidf_(sg[(1 * 16 + m) * 32 + c]);
      const float gt = tanhf    (sg[(2 * 16 + m) * 32 + c]);
      const float ot = sigmoidf_(sg[(3 * 16 + m) * 32 + c]);
      const float cold = sc[m * 32 + c];
      const float cnew = ft * cold + it * gt;
      const float preh = ot * tanhf(cold);     // reference uses OLD c here
      sc[m * 32 + c] = cnew;
      const size_t oidx = ((size_t)(r0 + m) * S_ + t) * H_ + j0 + c;
      cseq[oidx] = cnew;
      prehbuf[(r0 + m) * H_ + j0 + c] = preh;
      if (last) cT[(r0 + m) * H_ + j0 + c] = cnew;
    }
    rg_barrier(cnt, epoch, ++target);          // pre_h visible in row-group

    // -------- stage pre_h block [16 x 512] into LDS --------
    for (int i = tid * 4; i < 16 * 512; i += 128 * 4) {
      const int r = i >> 9, c = i & 511;
      const float4 v = *(const float4*)&prehbuf[(r0 + r) * H_ + c];
      float* d = &sh[r * 516 + c];
      d[0] = v.x; d[1] = v.y; d[2] = v.z; d[3] = v.w;
    }
    __syncthreads();

    // -------- phase 2: params = pre_h @ h_reparam (B from LDS, b64 loads) ----
    // wave quadrants over local cols: 0..15 mu-lo, 16..31 mu-hi, 32..47 std-lo,
    // 48..63 std-hi.
    const int lcb = wave * 16;
    v8f pacc = {};
    for (int k0 = 0; k0 < H_; k0 += 4) {
      const int ka = k0 + 2 * lhi;
      v2f a = *(const v2f*)&sh[lrow * 516 + ka];
      v2f b = *(const v2f*)&shr[(lcb + lrow) * 516 + ka];
      pacc = WMMA_F32X4(a, b, pacc);
    }
    {
      const int part = (wave < 2) ? 0 : 1;
      const int c = (wave & 1) * 16 + lrow;
      for (int v = 0; v < 8; ++v) {
        const int m = v + 8 * lhi;
        sg[part * 512 + m * 32 + c] = pacc[v];
      }
    }
    __syncthreads();

    // -------- mu/std/h_new --------
    for (int u = tid; u < 512; u += 128) {
      const int m = u >> 5, c = u & 31;
      float mu = sg[m * 32 + c];
      mu = fminf(fmaxf(mu, 1e-6f), 1e6f);
      const float sd = fmaxf(softplusf_(sg[512 + m * 32 + c]), 1e-6f);
      const size_t oidx = ((size_t)(r0 + m) * S_ + t) * H_ + j0 + c;
      const float hn = fmaf(eps[oidx], sd, mu);
      hseq[oidx] = hn;
      mus[oidx]  = mu;
      stds[oidx] = sd;
      hbuf[(r0 + m) * H_ + j0 + c] = hn;
      if (last) hT[(r0 + m) * H_ + j0 + c] = hn;
    }
    rg_barrier(cnt, epoch, ++target);          // h_new visible in row-group
  }
}

// ---------------------------------------------------------------------------
extern "C" void kernel_launch(void* const* d_in, const int* in_sizes, int n_in,
                              void* d_out, int out_size, void* d_ws, size_t ws_size,
                              hipStream_t stream) {
  (void)in_sizes; (void)n_in; (void)out_size;
  const float* x    = (const float*)d_in[0];
  const float* wih  = (const float*)d_in[1];
  const float* whh  = (const float*)d_in[2];
  const float* bias = (const float*)d_in[3];
  const float* hrep = (const float*)d_in[4];
  const float* eps  = (const float*)d_in[5];

  float* out  = (float*)d_out;
  const size_t NSH = (size_t)B_ * S_ * H_;
  float* hseq = out;
  float* cseq = out + NSH;
  float* mus  = out + 2 * NSH;
  float* stds = out + 3 * NSH;
  float* hT   = out + 4 * NSH;
  float* cT   = hT + (size_t)B_ * H_;

  // ws layout: h state | pre_h state | sync words(256) | (optional) x_proj
  float*    ws      = (float*)d_ws;
  float*    hbuf    = ws;
  float*    prehbuf = ws + (size_t)B_ * H_;
  unsigned* syncw   = (unsigned*)(ws + 2 * (size_t)B_ * H_);
  float*    xproj   = ws + 2 * (size_t)B_ * H_ + 256;
  const size_t need =
      (2 * (size_t)B_ * H_ + 256 + (size_t)B_ * S_ * G4_) * sizeof(float);
  const bool use_xproj = (ws_size >= need);

  init_state<<<dim3(64), dim3(256), 0, stream>>>(hbuf, syncw);

  if (use_xproj) {
    xproj_gemm<<<dim3((B_ * S_) / 16, G4_ / 64), dim3(128), 0, stream>>>(
        x, wih, bias, xproj);
    lstm_scan<true><<<dim3(NWG), dim3(128), 0, stream>>>(
        x, wih, whh, bias, hrep, eps, xproj, hbuf, prehbuf, syncw,
        hseq, cseq, mus, stds, hT, cT);
  } else {
    lstm_scan<false><<<dim3(NWG), dim3(128), 0, stream>>>(
        x, wih, whh, bias, hrep, eps, nullptr, hbuf, prehbuf, syncw,
        hseq, cseq, mus, stds, hT, cT);
  }
}